// GCN_7026566496715
// MI455X (gfx1250) — compile-verified
//
#include <hip/hip_runtime.h>
#include <math.h>

#define NODES 100000

typedef float v2f __attribute__((ext_vector_type(2)));
typedef float v8f __attribute__((ext_vector_type(8)));

// ---------------- degree / normalization ----------------
__global__ void k_deg_init(float* __restrict__ deg) {
  int i = blockIdx.x * blockDim.x + threadIdx.x;
  if (i < NODES) deg[i] = 1.0f;  // self-loop contribution
}

__global__ void k_deg_acc(const int* __restrict__ dst, float* __restrict__ deg, int E) {
  int e = blockIdx.x * blockDim.x + threadIdx.x;
  if (e < E) atomicAdd(&deg[dst[e]], 1.0f);
}

__global__ void k_rsqrt(float* __restrict__ deg) {
  int i = blockIdx.x * blockDim.x + threadIdx.x;
  if (i < NODES) deg[i] = rsqrtf(deg[i]);  // deg >= 1 always (self-loops)
}

// ---------------- GEMM1: g1 = (x @ W1) * dis[row]  (100000x512 @ 512x16) ----
// One wave per 16-row tile. V_WMMA_F32_16X16X4_F32, K stepped by 4.
// A 16x4 f32 layout: lanes 0-15 M=0..15; VGPR0 = K (lanes<16) / K+2 (lanes>=16),
// VGPR1 = K+1 / K+3.  B 4x16: same K split across lane halves, N = lane&15.
__global__ void k_gemm1_wmma(const float* __restrict__ x, const float* __restrict__ W1,
                             const float* __restrict__ dis, float* __restrict__ g1) {
  const int gwave = (blockIdx.x * blockDim.x + threadIdx.x) >> 5;
  const int lane  = threadIdx.x & 31;
  if (gwave >= NODES / 16) return;  // wave-uniform guard: EXEC all-1s for WMMA
  const int m     = lane & 15;
  const int hi    = lane >> 4;   // 0 or 1
  const int khalf = hi << 1;     // 0 or 2
  const int row   = gwave * 16 + m;
  const float* a_ptr = x  + (size_t)row * 512 + khalf;
  const float* b_ptr = W1 + (size_t)khalf * 16 + m;
  v8f c = {0.f, 0.f, 0.f, 0.f, 0.f, 0.f, 0.f, 0.f};
#pragma unroll 8
  for (int k = 0; k < 512; k += 4) {
    v2f a, b;
    a.x = a_ptr[k];
    a.y = a_ptr[k + 1];
    b.x = b_ptr[k * 16];
    b.y = b_ptr[k * 16 + 16];
    c = __builtin_amdgcn_wmma_f32_16x16x4_f32(false, a, false, b, (short)0, c,
                                              false, false);
  }
  // C/D: VGPR r, lanes 0-15 -> M=r, lanes 16-31 -> M=r+8; N = lane&15.
  const int mhi = hi << 3;
#pragma unroll
  for (int r = 0; r < 8; ++r) {
    int rr = gwave * 16 + r + mhi;
    g1[(size_t)rr * 16 + m] = c[r] * dis[rr];  // fold dis[src] into source features
  }
}

// ---------------- GEMM2: g2 = (h1 @ W2) * dis[row]  (100000x16 @ 16x64) -----
__global__ void k_gemm2_wmma(const float* __restrict__ h1, const float* __restrict__ W2,
                             const float* __restrict__ dis, float* __restrict__ g2) {
  const int gwave = (blockIdx.x * blockDim.x + threadIdx.x) >> 5;
  const int lane  = threadIdx.x & 31;
  if (gwave >= (NODES / 16) * 4) return;
  const int mtile = gwave >> 2;
  const int ntile = gwave & 3;
  const int m     = lane & 15;
  const int hi    = lane >> 4;
  const int khalf = hi << 1;
  const int row   = mtile * 16 + m;
  const float* a_ptr = h1 + (size_t)row * 16 + khalf;
  const float* b_ptr = W2 + (size_t)khalf * 64 + ntile * 16 + m;
  v8f c = {0.f, 0.f, 0.f, 0.f, 0.f, 0.f, 0.f, 0.f};
#pragma unroll
  for (int k = 0; k < 16; k += 4) {
    v2f a, b;
    a.x = a_ptr[k];
    a.y = a_ptr[k + 1];
    b.x = b_ptr[k * 64];
    b.y = b_ptr[k * 64 + 64];
    c = __builtin_amdgcn_wmma_f32_16x16x4_f32(false, a, false, b, (short)0, c,
                                              false, false);
  }
  const int mhi = hi << 3;
#pragma unroll
  for (int r = 0; r < 8; ++r) {
    int rr = mtile * 16 + r + mhi;
    g2[(size_t)rr * 64 + ntile * 16 + m] = c[r] * dis[rr];
  }
}

// ---------------- layer-1 propagation ----------------
__global__ void k_agg1_init(const float* __restrict__ g1, const float* __restrict__ dis,
                            float* __restrict__ agg1) {
  int t = blockIdx.x * blockDim.x + threadIdx.x;
  if (t < NODES * 16) agg1[t] = g1[t] * dis[t >> 4];  // self-loop term g1[i]*dis[i]
}

__global__ void k_msg1(const int* __restrict__ src, const int* __restrict__ dst,
                       const float* __restrict__ g1, const float* __restrict__ dis,
                       float* __restrict__ agg1, int E) {
  int e = blockIdx.x * blockDim.x + threadIdx.x;
  if (e >= E) return;
  int s = src[e], d = dst[e];
  float nd = dis[d];
  const float4* p = reinterpret_cast<const float4*>(g1 + (size_t)s * 16);
  float* q = agg1 + (size_t)d * 16;
  float4 v0 = p[0], v1 = p[1], v2 = p[2], v3 = p[3];
  atomicAdd(q + 0,  v0.x * nd); atomicAdd(q + 1,  v0.y * nd);
  atomicAdd(q + 2,  v0.z * nd); atomicAdd(q + 3,  v0.w * nd);
  atomicAdd(q + 4,  v1.x * nd); atomicAdd(q + 5,  v1.y * nd);
  atomicAdd(q + 6,  v1.z * nd); atomicAdd(q + 7,  v1.w * nd);
  atomicAdd(q + 8,  v2.x * nd); atomicAdd(q + 9,  v2.y * nd);
  atomicAdd(q + 10, v2.z * nd); atomicAdd(q + 11, v2.w * nd);
  atomicAdd(q + 12, v3.x * nd); atomicAdd(q + 13, v3.y * nd);
  atomicAdd(q + 14, v3.z * nd); atomicAdd(q + 15, v3.w * nd);
}

__global__ void k_relu(const float* __restrict__ agg1, const float* __restrict__ b1,
                       float* __restrict__ h1) {
  int t = blockIdx.x * blockDim.x + threadIdx.x;
  if (t < NODES * 16) h1[t] = fmaxf(agg1[t] + b1[t & 15], 0.0f);
}

// ---------------- layer-2 propagation (accumulate directly into d_out) -----
__global__ void k_out_init(const float* __restrict__ g2, const float* __restrict__ dis,
                           float* __restrict__ out) {
  int t = blockIdx.x * blockDim.x + threadIdx.x;
  if (t < NODES * 64) out[t] = g2[t] * dis[t >> 6];  // self-loop term
}

__global__ void k_msg2(const int* __restrict__ src, const int* __restrict__ dst,
                       const float* __restrict__ g2, const float* __restrict__ dis,
                       float* __restrict__ out, int E4) {
  int t = blockIdx.x * blockDim.x + threadIdx.x;
  if (t >= E4) return;
  int e = t >> 2, qd = (t & 3) * 16;
  int s = src[e], d = dst[e];
  float nd = dis[d];
  const float4* p = reinterpret_cast<const float4*>(g2 + (size_t)s * 64 + qd);
  float* q = out + (size_t)d * 64 + qd;
  float4 v0 = p[0], v1 = p[1], v2 = p[2], v3 = p[3];
  atomicAdd(q + 0,  v0.x * nd); atomicAdd(q + 1,  v0.y * nd);
  atomicAdd(q + 2,  v0.z * nd); atomicAdd(q + 3,  v0.w * nd);
  atomicAdd(q + 4,  v1.x * nd); atomicAdd(q + 5,  v1.y * nd);
  atomicAdd(q + 6,  v1.z * nd); atomicAdd(q + 7,  v1.w * nd);
  atomicAdd(q + 8,  v2.x * nd); atomicAdd(q + 9,  v2.y * nd);
  atomicAdd(q + 10, v2.z * nd); atomicAdd(q + 11, v2.w * nd);
  atomicAdd(q + 12, v3.x * nd); atomicAdd(q + 13, v3.y * nd);
  atomicAdd(q + 14, v3.z * nd); atomicAdd(q + 15, v3.w * nd);
}

// ---------------- final: out = log_softmax(out + b2), one wave32 per node ---
__global__ void k_logsoftmax(float* __restrict__ out, const float* __restrict__ b2) {
  int gt   = blockIdx.x * blockDim.x + threadIdx.x;
  int node = gt >> 5;
  int lane = gt & 31;
  if (node >= NODES) return;
  float* row = out + (size_t)node * 64;
  float v0 = row[lane]      + b2[lane];
  float v1 = row[lane + 32] + b2[lane + 32];
  float m = fmaxf(v0, v1);
#pragma unroll
  for (int off = 16; off > 0; off >>= 1) m = fmaxf(m, __shfl_xor(m, off, 32));
  float s = expf(v0 - m) + expf(v1 - m);
#pragma unroll
  for (int off = 16; off > 0; off >>= 1) s += __shfl_xor(s, off, 32);
  float lse = m + logf(s);
  row[lane]      = v0 - lse;
  row[lane + 32] = v1 - lse;
}

extern "C" void kernel_launch(void* const* d_in, const int* in_sizes, int n_in,
                              void* d_out, int out_size, void* d_ws, size_t ws_size,
                              hipStream_t stream) {
  const float* x  = (const float*)d_in[0];
  const int*   ei = (const int*)d_in[1];   // harness contract: integer -> const int*
  const float* W1 = (const float*)d_in[2];
  const float* b1 = (const float*)d_in[3];
  const float* W2 = (const float*)d_in[4];
  const float* b2 = (const float*)d_in[5];
  float* out = (float*)d_out;

  const int E = in_sizes[1] / 2;
  const int* src = ei;
  const int* dst = ei + E;

  char* ws = (char*)d_ws;
  float* dis  = (float*)(ws);               // NODES floats         (0.4 MB)
  float* g1   = (float*)(ws + (1u  << 20)); // NODES*16 floats      (6.4 MB)
  float* agg1 = (float*)(ws + (8u  << 20)); // NODES*16 floats      (6.4 MB)
  float* g2   = (float*)(ws + (16u << 20)); // NODES*64 floats      (25.6 MB)

  const int B = 256;
  k_deg_init <<<(NODES + B - 1) / B, B, 0, stream>>>(dis);
  k_deg_acc  <<<(E + B - 1) / B, B, 0, stream>>>(dst, dis, E);
  k_rsqrt    <<<(NODES + B - 1) / B, B, 0, stream>>>(dis);

  k_gemm1_wmma<<<((NODES / 16) * 32 + B - 1) / B, B, 0, stream>>>(x, W1, dis, g1);
  k_agg1_init <<<(NODES * 16 + B - 1) / B, B, 0, stream>>>(g1, dis, agg1);
  k_msg1      <<<(E + B - 1) / B, B, 0, stream>>>(src, dst, g1, dis, agg1, E);
  k_relu      <<<(NODES * 16 + B - 1) / B, B, 0, stream>>>(agg1, b1, g1); // h1 -> g1 buf

  k_gemm2_wmma<<<((NODES / 16) * 4 * 32 + B - 1) / B, B, 0, stream>>>(g1, W2, dis, g2);
  k_out_init  <<<(NODES * 64 + B - 1) / B, B, 0, stream>>>(g2, dis, out);
  k_msg2      <<<(E * 4 + B - 1) / B, B, 0, stream>>>(src, dst, g2, dis, out, E * 4);

  k_logsoftmax<<<(NODES * 32 + B - 1) / B, B, 0, stream>>>(out, b2);
}